// DriftingLoss_55422257987621
// MI455X (gfx1250) — compile-verified
//
#include <hip/hip_runtime.h>
#include <hip/hip_bf16.h>
#include <math.h>

typedef __attribute__((ext_vector_type(2))) float v2f;
typedef __attribute__((ext_vector_type(8))) float v8f;

#define NN 4096
#define DD 256
#define EPSV 1e-8f

static __device__ __forceinline__ v8f wmma4(v2f a, v2f b, v8f c) {
    // V_WMMA_F32_16X16X4_F32 : D = A(16x4) * B(4x16) + C(16x16), all fp32
    return __builtin_amdgcn_wmma_f32_16x16x4_f32(
        /*neg_a=*/false, a, /*neg_b=*/false, b,
        /*c_mod=*/(short)0, c, /*reuse_a=*/false, /*reuse_b=*/false);
}

// ------------------------------------------- per-row ||y_j||^2 (no atomics)
__global__ void k_rownorm(const float* __restrict__ yp, const float* __restrict__ yn,
                          float* __restrict__ yyp, float* __restrict__ yyn) {
    const bool neg = (blockIdx.x >= NN);
    const int row  = neg ? (blockIdx.x - NN) : blockIdx.x;
    const float* y = neg ? yn : yp;
    float v = y[(size_t)row * DD + threadIdx.x];
    float s = v * v;
    __shared__ float red[8];
#pragma unroll
    for (int o = 16; o > 0; o >>= 1) s += __shfl_down(s, o);
    if ((threadIdx.x & 31) == 0) red[threadIdx.x >> 5] = s;
    __syncthreads();
    if (threadIdx.x == 0) {
        float t = 0.0f;
        for (int q = 0; q < 8; q++) t += red[q];
        (neg ? yyn : yyp)[row] = t;
    }
}

// --------------------- scale^2 = mean(ref^2)+eps, deterministic single block
__global__ void k_scale(const float* __restrict__ yyp, const float* __restrict__ yyn,
                        float* __restrict__ acc) {
    __shared__ float red[8];
    float s = 0.0f;
    for (int j = threadIdx.x; j < NN; j += 256) s += yyp[j];
    for (int j = threadIdx.x; j < NN; j += 256) s += yyn[j];
#pragma unroll
    for (int o = 16; o > 0; o >>= 1) s += __shfl_down(s, o);
    if ((threadIdx.x & 31) == 0) red[threadIdx.x >> 5] = s;
    __syncthreads();
    if (threadIdx.x == 0) {
        float t = 0.0f;
        for (int q = 0; q < 8; q++) t += red[q];
        acc[1] = t / (2.0f * NN * DD) + EPSV;
    }
}

// ---------------------------------------------- S = 2 * x @ y^T - ||y_j||^2
// 1x2 register blocking: each wave computes two adjacent 16x16 j-tiles sharing
// the A fragment -> two independent WMMA chains, half the A traffic.
__global__ void k_sgemm(const float* __restrict__ x, const float* __restrict__ y,
                        const float* __restrict__ yy, float* __restrict__ S) {
    const int lane = threadIdx.x & 31;
    const int wave = threadIdx.x >> 5;
    const int t  = blockIdx.x * 8 + wave;   // 32768 tile-pairs
    const int it = t >> 7;                  // 128 pairs per i-tile row
    const int jp = t & 127;
    const int i0 = it * 16, j0 = jp * 32;
    const int lm = lane & 15;               // A: M row / B: N col for this lane
    const int kh = lane >> 4;               // k-half within fragment

    const float* xrow  = x + (size_t)(i0 + lm) * DD + 2 * kh;
    const float* yrow0 = y + (size_t)(j0 + lm) * DD + 2 * kh;
    const float* yrow1 = y + (size_t)(j0 + 16 + lm) * DD + 2 * kh;

    v8f c0 = {}, c1 = {};
#pragma unroll 8
    for (int k = 0; k < DD; k += 4) {
        v2f a  = *(const v2f*)(xrow + k);
        v2f b0 = *(const v2f*)(yrow0 + k);
        v2f b1 = *(const v2f*)(yrow1 + k);
        c0 = wmma4(a, b0, c0);
        c1 = wmma4(a, b1, c1);
    }
    // C layout: lanes<16 -> n=lane, m=r ; lanes>=16 -> n=lane-16, m=r+8
    const int n  = lane & 15;
    const int mb = (lane >> 4) * 8;
    const float yy0 = yy[j0 + n];
    const float yy1 = yy[j0 + 16 + n];
#pragma unroll
    for (int r = 0; r < 8; r++) {
        S[(size_t)(i0 + mb + r) * NN + j0 + n]      = 2.0f * c0[r] - yy0;
        S[(size_t)(i0 + mb + r) * NN + j0 + 16 + n] = 2.0f * c1[r] - yy1;
    }
}

// ----------------------------- per-row softmax stats: max + 3 exp-sum inverses
// rs[i*8] = {m_pos, m_neg, 1/l_pos[0..2], 1/l_neg[0..2]}
__global__ void k_rowstats(const float* __restrict__ Sp, const float* __restrict__ Sn,
                           const float* __restrict__ acc, float* __restrict__ rs) {
    const int i = blockIdx.x;
    const float scl2 = acc[1];
    const float r0 = 1.0f / (scl2 * 0.02f);
    const float r1 = 1.0f / (scl2 * 0.05f);
    const float r2 = 1.0f / (scl2 * 0.20f);
    __shared__ float redm[8];
    __shared__ float reds[3][8];
    __shared__ float bmax;
    for (int half = 0; half < 2; half++) {
        const float* S = (half ? Sn : Sp) + (size_t)i * NN;
        float mx = -3.4e38f;
        for (int j = threadIdx.x; j < NN; j += 256) mx = fmaxf(mx, S[j]);
#pragma unroll
        for (int o = 16; o > 0; o >>= 1) mx = fmaxf(mx, __shfl_down(mx, o));
        if ((threadIdx.x & 31) == 0) redm[threadIdx.x >> 5] = mx;
        __syncthreads();
        if (threadIdx.x == 0) {
            float mm = redm[0];
            for (int q = 1; q < 8; q++) mm = fmaxf(mm, redm[q]);
            bmax = mm;
        }
        __syncthreads();
        const float m = bmax;
        float s0 = 0.0f, s1 = 0.0f, s2 = 0.0f;
        for (int j = threadIdx.x; j < NN; j += 256) {
            float d = S[j] - m;
            s0 += expf(d * r0); s1 += expf(d * r1); s2 += expf(d * r2);
        }
#pragma unroll
        for (int o = 16; o > 0; o >>= 1) {
            s0 += __shfl_down(s0, o); s1 += __shfl_down(s1, o); s2 += __shfl_down(s2, o);
        }
        if ((threadIdx.x & 31) == 0) {
            int w = threadIdx.x >> 5;
            reds[0][w] = s0; reds[1][w] = s1; reds[2][w] = s2;
        }
        __syncthreads();
        if (threadIdx.x == 0) {
            float t0 = 0.0f, t1 = 0.0f, t2 = 0.0f;
            for (int q = 0; q < 8; q++) { t0 += reds[0][q]; t1 += reds[1][q]; t2 += reds[2][q]; }
            rs[i * 8 + half]              = m;
            rs[i * 8 + 2 + half * 3 + 0]  = 1.0f / t0;
            rs[i * 8 + 2 + half * 3 + 1]  = 1.0f / t1;
            rs[i * 8 + 2 + half * 3 + 2]  = 1.0f / t2;
        }
        __syncthreads();
    }
}

// -------------------------------------------------- fused V-field computation
// grid (NN/16, 2), block 256 (8 waves). Each block owns a 16-row i-tile and a
// 128-column slice of D. Producer: wave w builds 6 probability tiles for
// j-subtile w into LDS (A-fragment layout => consumer reads are ds_load_b64).
// Consumer: 6 WMMA accumulators (3 taus x pos/neg) per wave over its 16 cols.
// Partial norm sums go to `part` (deterministic fixed-order reduction later).
__global__ void k_vfield(const float* __restrict__ Sp, const float* __restrict__ Sn,
                         const float* __restrict__ ypos, const float* __restrict__ yneg,
                         const float* __restrict__ rs, const float* __restrict__ acc,
                         float* __restrict__ part, float* __restrict__ Vout) {
    __shared__ v2f ldsP[8][6][4][32];   // 48 KB: [jtile][set][kstep][lane]
    __shared__ float redp[8][4];

    const int lane = threadIdx.x & 31;
    const int wave = threadIdx.x >> 5;
    const int i0 = blockIdx.x * 16;
    const int n0 = blockIdx.y * 128 + wave * 16;
    const int lm = lane & 15;
    const int kh = lane >> 4;

    const float scl2 = acc[1];
    const float r0 = 1.0f / (scl2 * 0.02f);
    const float r1 = 1.0f / (scl2 * 0.05f);
    const float r2 = 1.0f / (scl2 * 0.20f);

    const float* rsr = rs + (size_t)(i0 + lm) * 8;
    const float mp = rsr[0], mn = rsr[1];
    const float ip0 = rsr[2], ip1 = rsr[3], ip2 = rsr[4];
    const float in0 = rsr[5], in1 = rsr[6], in2 = rsr[7];

    const float* sprow = Sp + (size_t)(i0 + lm) * NN;
    const float* snrow = Sn + (size_t)(i0 + lm) * NN;
    const float* ypb = ypos + n0 + lm;
    const float* ynb = yneg + n0 + lm;

    v8f aP0 = {}, aP1 = {}, aP2 = {}, aN0 = {}, aN1 = {}, aN2 = {};

    for (int jb = 0; jb < NN; jb += 128) {
        // ---- producer: this wave's j-subtile -> 6 prob tiles in LDS
        const int jt = jb + wave * 16;
#pragma unroll
        for (int step = 0; step < 4; step++) {
            const int col = jt + step * 4 + 2 * kh;
            v2f sp = *(const v2f*)(sprow + col);
            v2f sn = *(const v2f*)(snrow + col);
            v2f p;
            p.x = expf((sp.x - mp) * r0) * ip0; p.y = expf((sp.y - mp) * r0) * ip0;
            ldsP[wave][0][step][lane] = p;
            p.x = expf((sp.x - mp) * r1) * ip1; p.y = expf((sp.y - mp) * r1) * ip1;
            ldsP[wave][1][step][lane] = p;
            p.x = expf((sp.x - mp) * r2) * ip2; p.y = expf((sp.y - mp) * r2) * ip2;
            ldsP[wave][2][step][lane] = p;
            p.x = expf((sn.x - mn) * r0) * in0; p.y = expf((sn.y - mn) * r0) * in0;
            ldsP[wave][3][step][lane] = p;
            p.x = expf((sn.x - mn) * r1) * in1; p.y = expf((sn.y - mn) * r1) * in1;
            ldsP[wave][4][step][lane] = p;
            p.x = expf((sn.x - mn) * r2) * in2; p.y = expf((sn.y - mn) * r2) * in2;
            ldsP[wave][5][step][lane] = p;
        }
        __syncthreads();
        // ---- consumer: 8 j-subtiles x 4 k-steps x 6 WMMA
        for (int t = 0; t < 8; t++) {
#pragma unroll
            for (int step = 0; step < 4; step++) {
                const int krow = jb + t * 16 + step * 4 + 2 * kh;
                v2f bp, bn;
                bp.x = ypb[(size_t)krow * DD];       bp.y = ypb[(size_t)(krow + 1) * DD];
                bn.x = ynb[(size_t)krow * DD];       bn.y = ynb[(size_t)(krow + 1) * DD];
                aP0 = wmma4(ldsP[t][0][step][lane], bp, aP0);
                aP1 = wmma4(ldsP[t][1][step][lane], bp, aP1);
                aP2 = wmma4(ldsP[t][2][step][lane], bp, aP2);
                aN0 = wmma4(ldsP[t][3][step][lane], bn, aN0);
                aN1 = wmma4(ldsP[t][4][step][lane], bn, aN1);
                aN2 = wmma4(ldsP[t][5][step][lane], bn, aN2);
            }
        }
        __syncthreads();
    }

    // ---- epilogue: V_tau = P_pos@y_pos - P_neg@y_neg (scale cancels)
    const int ncol = n0 + (lane & 15);
    const int mb = (lane >> 4) * 8;
    float pt0 = 0.0f, pt1 = 0.0f, pt2 = 0.0f, raw = 0.0f;
#pragma unroll
    for (int r = 0; r < 8; r++) {
        float v0  = aP0[r] - aN0[r];
        float v1  = aP1[r] - aN1[r];
        float v2_ = aP2[r] - aN2[r];
        float vc  = (v0 + v1 + v2_) * (1.0f / 3.0f);
        Vout[(size_t)(i0 + mb + r) * DD + ncol] = vc;
        pt0 += v0 * v0; pt1 += v1 * v1; pt2 += v2_ * v2_; raw += vc * vc;
    }
#pragma unroll
    for (int o = 16; o > 0; o >>= 1) {
        raw += __shfl_down(raw, o);
        pt0 += __shfl_down(pt0, o); pt1 += __shfl_down(pt1, o); pt2 += __shfl_down(pt2, o);
    }
    if (lane == 0) {
        redp[wave][0] = raw; redp[wave][1] = pt0; redp[wave][2] = pt1; redp[wave][3] = pt2;
    }
    __syncthreads();
    if (threadIdx.x == 0) {
        float t0 = 0.0f, t1 = 0.0f, t2 = 0.0f, t3 = 0.0f;
        for (int q = 0; q < 8; q++) {
            t0 += redp[q][0]; t1 += redp[q][1]; t2 += redp[q][2]; t3 += redp[q][3];
        }
        const int bid = blockIdx.x * 2 + blockIdx.y;   // 512 blocks
        part[bid * 4 + 0] = t0; part[bid * 4 + 1] = t1;
        part[bid * 4 + 2] = t2; part[bid * 4 + 3] = t3;
    }
}

// --------------------- scalars (deterministic fixed-order sum of 512 partials)
__global__ void k_finalize(const float* __restrict__ part, float* __restrict__ acc,
                           float* __restrict__ out) {
    __shared__ float red[8][4];
    float s0 = 0.0f, s1 = 0.0f, s2 = 0.0f, s3 = 0.0f;
    for (int b = threadIdx.x; b < 512; b += 256) {
        s0 += part[b * 4 + 0]; s1 += part[b * 4 + 1];
        s2 += part[b * 4 + 2]; s3 += part[b * 4 + 3];
    }
#pragma unroll
    for (int o = 16; o > 0; o >>= 1) {
        s0 += __shfl_down(s0, o); s1 += __shfl_down(s1, o);
        s2 += __shfl_down(s2, o); s3 += __shfl_down(s3, o);
    }
    if ((threadIdx.x & 31) == 0) {
        int w = threadIdx.x >> 5;
        red[w][0] = s0; red[w][1] = s1; red[w][2] = s2; red[w][3] = s3;
    }
    __syncthreads();
    if (threadIdx.x == 0) {
        float t0 = 0.0f, t1 = 0.0f, t2 = 0.0f, t3 = 0.0f;
        for (int q = 0; q < 8; q++) {
            t0 += red[q][0]; t1 += red[q][1]; t2 += red[q][2]; t3 += red[q][3];
        }
        float raw   = t0 / (float)NN;                 // mean_i sum_d Vc^2
        float lam   = sqrtf(raw / (float)DD + EPSV);  // lambda_V
        float drift = raw / (lam * lam);              // after normalization
        out[0] = drift / (float)DD;                   // loss
        out[1] = drift;                               // drift_norm
        out[2] = raw;                                 // raw_drift_norm
        out[3] = lam;                                 // lambda_V
        out[4] = t1 / (float)NN;                      // per_temp[0]
        out[5] = t2 / (float)NN;                      // per_temp[1]
        out[6] = t3 / (float)NN;                      // per_temp[2]
        acc[6] = 1.0f / lam;
    }
}

// ---------------------------------------------------------- V normalization
__global__ void k_vnorm(float* __restrict__ V, const float* __restrict__ acc) {
    size_t idx = (size_t)blockIdx.x * 256 + threadIdx.x;
    V[idx] *= acc[6];
}

extern "C" void kernel_launch(void* const* d_in, const int* in_sizes, int n_in,
                              void* d_out, int out_size, void* d_ws, size_t ws_size,
                              hipStream_t stream) {
    const float* x    = (const float*)d_in[0];
    const float* ypos = (const float*)d_in[1];
    const float* yneg = (const float*)d_in[2];
    float* out = (float*)d_out;

    // workspace layout (floats)
    float* Sp   = (float*)d_ws;
    float* Sn   = Sp  + (size_t)NN * NN;
    float* yyp  = Sn  + (size_t)NN * NN;
    float* yyn  = yyp + NN;
    float* rsb  = yyn + NN;
    float* acc  = rsb + (size_t)NN * 8;
    float* part = acc + 16;                 // 512 * 4 partial sums

    float* Vout = out + 7;

    k_rownorm<<<2 * NN, 256, 0, stream>>>(ypos, yneg, yyp, yyn);
    k_scale<<<1, 256, 0, stream>>>(yyp, yyn, acc);
    k_sgemm<<<4096, 256, 0, stream>>>(x, ypos, yyp, Sp);
    k_sgemm<<<4096, 256, 0, stream>>>(x, yneg, yyn, Sn);
    k_rowstats<<<NN, 256, 0, stream>>>(Sp, Sn, acc, rsb);
    k_vfield<<<dim3(NN / 16, 2), 256, 0, stream>>>(Sp, Sn, ypos, yneg, rsb, acc, part, Vout);
    k_finalize<<<1, 256, 0, stream>>>(part, acc, out);
    k_vnorm<<<(NN * DD) / 256, 256, 0, stream>>>(Vout, acc);
}